// CQCA_cfa_59115929862339
// MI455X (gfx1250) — compile-verified
//
#include <hip/hip_runtime.h>
#include <math.h>

// ---------------------------------------------------------------------------
// MI455X (gfx1250) implementation.
// Memory-bound workload (~49 GFLOP vs ~1.5-2 GB HBM traffic @ 23.3 TB/s) ->
// keep fp32, use V_WMMA_F32_16X16X4_F32 for the conv GEMMs (wave32 tiles:
// one wave = 16 output pixels x 16 output channels). Input halo + weight
// tiles are staged in LDS so the WMMA loop is branch-free: 4 ds_load + wmma.
// y1 (64MB) / y2 (128MB) stay resident in the 192MB L2.
// ---------------------------------------------------------------------------

typedef __attribute__((ext_vector_type(2))) float v2f;
typedef __attribute__((ext_vector_type(4))) float v4f;
typedef __attribute__((ext_vector_type(8))) float v8f;

#define WDIM   512
#define HDIM   512
#define PLANE  (WDIM * HDIM)          // 262144 floats per channel plane
#define NPTS   65536
#define BATCH  4
#define NBINS  2049                   // NCLUST + 1
#define RES_   0.16f
#define DBY_   40.96f
#define EPS_   1e-3f
#define DENS_T 100.0f

// ---------------------------------------------------------------------------
// 1) Per-batch label histogram: LDS bins + single global merge per block.
//    grid = (256, BATCH), block = 256  -> one point per thread.
// ---------------------------------------------------------------------------
__global__ void hist_kernel(const int* __restrict__ labels,
                            unsigned* __restrict__ counts) {
  __shared__ unsigned bins[NBINS];
  const int b = blockIdx.y;
  for (int i = threadIdx.x; i < NBINS; i += blockDim.x) bins[i] = 0;
  __syncthreads();
  const int idx = blockIdx.x * blockDim.x + threadIdx.x;
  const int l = labels[(long)b * NPTS + idx];      // l in [-1, 2047]
  atomicAdd(&bins[l + 1], 1u);                     // DS atomic
  __syncthreads();
  for (int i = threadIdx.x; i < NBINS; i += blockDim.x) {
    unsigned v = bins[i];
    if (v) atomicAdd(&counts[b * NBINS + i], v);   // global atomic merge
  }
}

// ---------------------------------------------------------------------------
// 2) Density relabel + BEV scatter (mode='drop' == bounds check; duplicate
//    indices are order-undefined in the reference, plain stores are fine).
//    bev layout: [b][c][by][bx], c-plane stride = PLANE.
// ---------------------------------------------------------------------------
__global__ void scatter_kernel(const float* __restrict__ points,
                               const int* __restrict__ labels,
                               const unsigned* __restrict__ counts,
                               float* __restrict__ bev) {
  const int b = blockIdx.y;
  const int i = blockIdx.x * blockDim.x + threadIdx.x;
  const float* p = points + ((long)b * NPTS + i) * 5;
  const float x  = p[1];
  const float yv = p[2];
  const float vv = p[4];
  const int l = labels[(long)b * NPTS + i];
  const float dens = (l != -1) ? (float)counts[b * NBINS + l + 1] : 0.0f;
  const int l2 = ((dens > DENS_T) && (l != -1)) ? -1 : l;
  const bool valid = (l2 > 0);
  int bx = (int)floorf(x / RES_) - 1;
  const int by = (int)floorf((yv + DBY_) / RES_) - 1;
  if (!valid) bx = WDIM;                            // forces drop
  if ((unsigned)bx < (unsigned)WDIM && (unsigned)by < (unsigned)HDIM) {
    float* base = bev + (long)b * 3 * PLANE + by * WDIM + bx;
    base[0 * PLANE] = vv;
    base[1 * PLANE] = dens;
    base[2 * PLANE] = (float)l2;
  }
}

// ---------------------------------------------------------------------------
// 3) 3x3 SAME conv as implicit GEMM with V_WMMA_F32_16X16X4_F32.
//    Block = 128 threads = 4 waves covering 64 consecutive output x of one
//    row; grid.x = BATCH*HDIM*(WDIM/64) = 16384, grid.y = COUT/16.
//
//    Staging (one barrier per block):
//      lds_in[ci][r][col] : CINP x 3 x 66-wide halo (cols padded to 68),
//                           boundary zeros resolved here -> branch-free loop.
//      lds_w [tap][ci][n] : 9 x CINP x 16 weight tile, shared by all waves.
//    CIN padded to CINP (multiple of 4) with zeros -> no K-tail guard.
//
//    WMMA fragments (wave32):
//      A (16x4): lane m=lane&15 holds pixel, halves hold K {0,1}/{2,3}.
//      B (4x16): lane n=lane&15 holds out-channel, same K split.
//      D (16x16): lane holds 8 consecutive-x pixels of channel n.
// ---------------------------------------------------------------------------
template <int CIN, int COUT>
__global__ __launch_bounds__(128)
void conv3x3_wmma_kernel(const float* __restrict__ in,
                         const float* __restrict__ w,
                         float* __restrict__ out) {
  constexpr int CINP = (CIN + 3) & ~3;     // 4, 16, 32
  constexpr int K    = CIN * 9;
  constexpr int LDSW = 68;                 // halo row pitch (66 used)

  __shared__ float lds_in[CINP * 3 * LDSW];
  __shared__ float lds_w[9 * CINP * 16];

  const int xB  = blockIdx.x & 7;                   // 8 x 64-wide tiles / row
  const int oy  = (blockIdx.x >> 3) & (HDIM - 1);
  const int b   = blockIdx.x >> 12;
  const int ocB = blockIdx.y * 16;
  const int x0  = xB * 64;

  const float* inB = in + (long)b * CIN * PLANE;

  // ---- stage input halo (zeros for padding / Cin padding) ----
  for (int idx = threadIdx.x; idx < CINP * 3 * 66; idx += 128) {
    const int col = idx % 66;
    const int t   = idx / 66;
    const int r   = t % 3;
    const int ci  = t / 3;
    const int iy  = oy + r - 1;
    const int ix  = x0 + col - 1;
    float v = 0.0f;
    if (ci < CIN && (unsigned)iy < (unsigned)HDIM && (unsigned)ix < (unsigned)WDIM)
      v = inB[((long)ci * HDIM + iy) * WDIM + ix];
    lds_in[(ci * 3 + r) * LDSW + col] = v;
  }
  // ---- stage weight tile [tap][ci][n] ----
  for (int idx = threadIdx.x; idx < 9 * CINP * 16; idx += 128) {
    const int n   = idx & 15;
    const int t   = idx >> 4;
    const int ci  = t % CINP;
    const int tap = t / CINP;
    float v = (ci < CIN) ? w[(long)(ocB + n) * K + ci * 9 + tap] : 0.0f;
    lds_w[(tap * CINP + ci) * 16 + n] = v;
  }
  __syncthreads();

  const int lane  = threadIdx.x & 31;
  const int waveX = (threadIdx.x >> 5) << 4;  // wave's 16-pixel sub-tile
  const int mn    = lane & 15;                // pixel for A, channel for B/D
  const int kh    = (lane >> 4) << 1;         // 0 or 2

  v8f acc = {};
#pragma unroll
  for (int r = 0; r < 3; ++r) {
#pragma unroll
    for (int s = 0; s < 3; ++s) {
      const int col = waveX + mn + s;         // ix = x0 + waveX + mn + s - 1
      const int tap = r * 3 + s;
      const float* aP = &lds_in[r * LDSW + col];       // + ci*3*LDSW
      const float* bP = &lds_w[tap * CINP * 16 + mn];  // + ci*16
#pragma unroll
      for (int ci0 = 0; ci0 < CINP; ci0 += 4) {
        v2f a, bf;
        a.x  = aP[(ci0 + kh) * 3 * LDSW];
        a.y  = aP[(ci0 + kh + 1) * 3 * LDSW];
        bf.x = bP[(ci0 + kh) * 16];
        bf.y = bP[(ci0 + kh + 1) * 16];
        acc = __builtin_amdgcn_wmma_f32_16x16x4_f32(
            /*neg_a=*/false, a, /*neg_b=*/false, bf,
            /*c_mod=*/(short)0, acc, /*reuse_a=*/false, /*reuse_b=*/false);
      }
    }
  }

  // Store: lane holds channel (ocB+mn), pixels x0+waveX + 8*(lane>=16) + [0..7].
  float* outP = out + (((long)b * COUT + (ocB + mn)) * HDIM + oy) * WDIM +
                x0 + waveX + ((lane >> 4) << 3);
  v4f lo = {acc[0], acc[1], acc[2], acc[3]};
  v4f hi = {acc[4], acc[5], acc[6], acc[7]};
  *(v4f*)(outP)     = lo;
  *(v4f*)(outP + 4) = hi;
}

// ---------------------------------------------------------------------------
// 4) BatchNorm stats: one block per (b,c) plane; deterministic tree reduce.
//    grid.x = BATCH*C, block = 256.
// ---------------------------------------------------------------------------
__global__ void bn_stats_kernel(const float* __restrict__ y,
                                float* __restrict__ planeSum,
                                float* __restrict__ planeSqs) {
  const int bc = blockIdx.x;
  const v4f* p = (const v4f*)(y + (long)bc * PLANE);
  float s = 0.0f, ss = 0.0f;
  for (int i = threadIdx.x; i < PLANE / 4; i += blockDim.x) {
    v4f v = p[i];
    s  += v.x + v.y + v.z + v.w;
    ss += v.x * v.x + v.y * v.y + v.z * v.z + v.w * v.w;
  }
  __shared__ float ls[256], lss[256];
  ls[threadIdx.x] = s; lss[threadIdx.x] = ss;
  __syncthreads();
  for (int o = 128; o > 0; o >>= 1) {
    if (threadIdx.x < o) {
      ls[threadIdx.x]  += ls[threadIdx.x + o];
      lss[threadIdx.x] += lss[threadIdx.x + o];
    }
    __syncthreads();
  }
  if (threadIdx.x == 0) { planeSum[bc] = ls[0]; planeSqs[bc] = lss[0]; }
}

__global__ void bn_finalize_kernel(const float* __restrict__ planeSum,
                                   const float* __restrict__ planeSqs,
                                   const float* __restrict__ g,
                                   const float* __restrict__ beta,
                                   float* __restrict__ scale,
                                   float* __restrict__ shift, int C) {
  const int c = threadIdx.x;
  if (c >= C) return;
  float s = 0.0f, ss = 0.0f;
  for (int b = 0; b < BATCH; ++b) {
    s  += planeSum[b * C + c];
    ss += planeSqs[b * C + c];
  }
  const float n    = (float)BATCH * PLANE;
  const float mean = s / n;
  const float var  = ss / n - mean * mean;   // population var, ddof=0
  const float sc   = g[c] * rsqrtf(var + EPS_);
  scale[c] = sc;
  shift[c] = beta[c] - mean * sc;
}

__global__ void bn_apply_kernel(float* __restrict__ y,
                                const float* __restrict__ scale,
                                const float* __restrict__ shift,
                                int C, long total4) {
  v4f* py = (v4f*)y;
  for (long i = (long)blockIdx.x * blockDim.x + threadIdx.x; i < total4;
       i += (long)gridDim.x * blockDim.x) {
    const int c = (int)((i >> 16) % C);      // PLANE/4 = 2^16 v4f per plane
    const float sc = scale[c], sh = shift[c];
    v4f v = py[i];
    v.x = fmaxf(v.x * sc + sh, 0.0f);
    v.y = fmaxf(v.y * sc + sh, 0.0f);
    v.z = fmaxf(v.z * sc + sh, 0.0f);
    v.w = fmaxf(v.w * sc + sh, 0.0f);
    py[i] = v;
  }
}

// ---------------------------------------------------------------------------
// Host launch
// ---------------------------------------------------------------------------
extern "C" void kernel_launch(void* const* d_in, const int* in_sizes, int n_in,
                              void* d_out, int out_size, void* d_ws,
                              size_t ws_size, hipStream_t stream) {
  (void)in_sizes; (void)n_in; (void)out_size; (void)ws_size;

  const float* points = (const float*)d_in[0];
  const int*   labels = (const int*)d_in[1];
  const float* w1 = (const float*)d_in[2];
  const float* g1 = (const float*)d_in[3];
  const float* b1 = (const float*)d_in[4];
  const float* w2 = (const float*)d_in[5];
  const float* g2 = (const float*)d_in[6];
  const float* b2 = (const float*)d_in[7];
  const float* w3 = (const float*)d_in[8];
  const float* g3 = (const float*)d_in[9];
  const float* b3 = (const float*)d_in[10];

  char* ws = (char*)d_ws;
  const size_t bevBytes = (size_t)BATCH * 3  * PLANE * 4;  //  12 MB
  const size_t y1Bytes  = (size_t)BATCH * 16 * PLANE * 4;  //  64 MB
  const size_t y2Bytes  = (size_t)BATCH * 32 * PLANE * 4;  // 128 MB
  float* bev = (float*)ws;
  float* y1  = (float*)(ws + bevBytes);
  float* y2  = (float*)(ws + bevBytes + y1Bytes);
  size_t off = bevBytes + y1Bytes + y2Bytes;
  off = (off + 255) & ~(size_t)255;
  unsigned* counts = (unsigned*)(ws + off); off += (size_t)BATCH * NBINS * 4;
  off = (off + 255) & ~(size_t)255;
  float* planeSum = (float*)(ws + off); off += (size_t)BATCH * 64 * 4;
  float* planeSqs = (float*)(ws + off); off += (size_t)BATCH * 64 * 4;
  float* scale    = (float*)(ws + off); off += 64 * 4;
  float* shift    = (float*)(ws + off); off += 64 * 4;
  float* y3 = (float*)d_out;

  // BEV build
  hipMemsetAsync(counts, 0, (size_t)BATCH * NBINS * 4, stream);
  hipMemsetAsync(bev, 0, bevBytes, stream);
  hist_kernel<<<dim3(NPTS / 256, BATCH), 256, 0, stream>>>(labels, counts);
  scatter_kernel<<<dim3(NPTS / 256, BATCH), 256, 0, stream>>>(points, labels,
                                                              counts, bev);
  const int convBlocksX = BATCH * HDIM * (WDIM / 64);  // 16384

  // Layer 1: 3 -> 16
  conv3x3_wmma_kernel<3, 16>
      <<<dim3(convBlocksX, 1), 128, 0, stream>>>(bev, w1, y1);
  bn_stats_kernel<<<BATCH * 16, 256, 0, stream>>>(y1, planeSum, planeSqs);
  bn_finalize_kernel<<<1, 64, 0, stream>>>(planeSum, planeSqs, g1, b1,
                                           scale, shift, 16);
  bn_apply_kernel<<<4096, 256, 0, stream>>>(y1, scale, shift, 16,
                                            (long)BATCH * 16 * PLANE / 4);
  // Layer 2: 16 -> 32
  conv3x3_wmma_kernel<16, 32>
      <<<dim3(convBlocksX, 2), 128, 0, stream>>>(y1, w2, y2);
  bn_stats_kernel<<<BATCH * 32, 256, 0, stream>>>(y2, planeSum, planeSqs);
  bn_finalize_kernel<<<1, 64, 0, stream>>>(planeSum, planeSqs, g2, b2,
                                           scale, shift, 32);
  bn_apply_kernel<<<8192, 256, 0, stream>>>(y2, scale, shift, 32,
                                            (long)BATCH * 32 * PLANE / 4);
  // Layer 3: 32 -> 64 (written straight into d_out, normalized in place)
  conv3x3_wmma_kernel<32, 64>
      <<<dim3(convBlocksX, 4), 128, 0, stream>>>(y2, w3, y3);
  bn_stats_kernel<<<BATCH * 64, 256, 0, stream>>>(y3, planeSum, planeSqs);
  bn_finalize_kernel<<<1, 64, 0, stream>>>(planeSum, planeSqs, g3, b3,
                                           scale, shift, 64);
  bn_apply_kernel<<<16384, 256, 0, stream>>>(y3, scale, shift, 64,
                                             (long)BATCH * 64 * PLANE / 4);
}